// GraphAttention_23519240913516
// MI455X (gfx1250) — compile-verified
//
#include <hip/hip_runtime.h>

// CDNA5 / gfx1250 vector types for WMMA operands.
typedef __attribute__((ext_vector_type(2))) float v2f;   // A/B of V_WMMA_F32_16X16X4_F32 (64 elems / 32 lanes)
typedef __attribute__((ext_vector_type(8))) float v8f;   // C/D 16x16 f32 accumulator

// Problem constants (from reference: x is (8, 64, 4096, 20) f32)
#define BD   8
#define CD   64
#define ND   4096
#define KD   20
#define NB   16                 // n values per workgroup
#define PAIRS (NB * KD)         // 320 (n,k) pairs per tile
#define GROUPS (PAIRS / 16)     // 20 WMMA column groups
#define THREADS 128             // 4 wave32

__global__ __launch_bounds__(THREADS)
void graph_attn_wmma_kernel(const float* __restrict__ x,
                            const float* __restrict__ W,     // (1, 2C): [0..63]=w_nbr, [64..127]=w_ctr
                            const float* __restrict__ bias,  // (1,)
                            float* __restrict__ out)         // (B, C, N, 1)
{
    // LDS tile: x[b, c, n0:n0+NB, :] laid out [c][n][k] -> 64*320 f32 = 80 KB (CDNA5: 320 KB/WGP)
    __shared__ float ldsX[CD * PAIRS];
    __shared__ float ldsScores[PAIRS];
    __shared__ float ldsAttn[PAIRS];
    __shared__ float ldsExtra[NB];

    const int wg  = blockIdx.x;             // B * (N/NB) = 2048 workgroups
    const int b   = wg / (ND / NB);
    const int n0  = (wg % (ND / NB)) * NB;
    const int tid = threadIdx.x;

    const float* xb = x + ((size_t)b * CD * ND + n0) * KD;   // &x[b, 0, n0, 0]

    // ---------- Phase 1: stage tile into LDS (read x exactly once, float4 coalesced) ----------
    {
        const int nf4 = (CD * PAIRS) / 4;   // 5120 float4 per tile
        for (int i = tid; i < nf4; i += THREADS) {
            const int c = i / (PAIRS / 4);
            const int r = i % (PAIRS / 4);
            const float4 v = *reinterpret_cast<const float4*>(xb + (size_t)c * ND * KD + r * 4);
            *reinterpret_cast<float4*>(&ldsX[c * PAIRS + r * 4]) = v;
        }
    }
    __syncthreads();

    // ---------- Phase 2a: scores via V_WMMA_F32_16X16X4_F32 (fp32, reference-exact precision) ----------
    // GEMM view: D[m, p] = sum_c A[m, c] * B[c, p], contraction over c in chunks of 4.
    // A: every row = w_nbr[c0..c0+3]  (so all 16 D rows hold the correct scores; pick row 0).
    // B: 4x16 slice of the LDS tile over pair-columns p = g*16 + (lane & 15).
    // VGPR layout (ISA 7.12.2): lanes 0-15 carry K=0 (v[0]) / K=1 (v[1]); lanes 16-31 carry K=2 / K=3.
    const int  lane = tid & 31;
    const int  wave = tid >> 5;
    const bool hi   = lane >= 16;
    for (int gi = 0; gi < GROUPS / 4; ++gi) {        // 5 column-groups per wave, wave-uniform (EXEC all-1s)
        const int g = wave * (GROUPS / 4) + gi;
        const int p = g * 16 + (lane & 15);
        v8f acc = {};
        for (int c0 = 0; c0 < CD; c0 += 4) {
            v2f a, bm;
            a[0]  = W[c0 + (hi ? 2 : 0)];            // uniform scalar loads of w_nbr
            a[1]  = W[c0 + (hi ? 3 : 1)];
            bm[0] = ldsX[(c0 + (hi ? 2 : 0)) * PAIRS + p];
            bm[1] = ldsX[(c0 + (hi ? 3 : 1)) * PAIRS + p];
            acc = __builtin_amdgcn_wmma_f32_16x16x4_f32(
                /*neg_a=*/false, a, /*neg_b=*/false, bm,
                /*c_mod=*/(short)0, acc, /*reuse_a=*/false, /*reuse_b=*/false);
        }
        if (!hi) ldsScores[p] = acc[0];              // D row 0: VGPR0, lanes 0-15
    }

    // ---------- Phase 2b: center term  extra[n] = b + sum_c w_ctr[c] * x[b,c,n,0] ----------
    if (tid < NB) {
        float e = bias[0];
        for (int c = 0; c < CD; ++c)
            e += W[CD + c] * ldsX[c * PAIRS + tid * KD];
        ldsExtra[tid] = e;
    }
    __syncthreads();

    // ---------- Phase 3: leaky_relu(0.2) + softmax over K=20, one thread per n ----------
    if (tid < NB) {
        float s[KD];
        const float e = ldsExtra[tid];
        float mx = -__builtin_inff();
        for (int k = 0; k < KD; ++k) {
            float v = ldsScores[tid * KD + k] + e;
            v = (v < 0.0f) ? 0.2f * v : v;           // leaky_relu, negative_slope = 0.2
            s[k] = v;
            mx = fmaxf(mx, v);
        }
        float sum = 0.0f;
        for (int k = 0; k < KD; ++k) { s[k] = __expf(s[k] - mx); sum += s[k]; }
        const float inv = 1.0f / sum;
        for (int k = 0; k < KD; ++k) ldsAttn[tid * KD + k] = s[k] * inv;
    }
    __syncthreads();

    // ---------- Phase 4: out[b,c,n] = sum_k x[b,c,n,k] * attn[n,k]; coalesced over n ----------
    {
        const int n  = tid & (NB - 1);       // threads 0..15 -> n 0..15 (contiguous stores)
        const int cb = tid >> 4;             // 0..7
        float at[KD];
        for (int k = 0; k < KD; ++k) at[k] = ldsAttn[n * KD + k];
        for (int j = 0; j < CD / 8; ++j) {
            const int c = cb * 8 + j;
            float acc = 0.0f;
            for (int k = 0; k < KD; ++k)
                acc = fmaf(ldsX[c * PAIRS + n * KD + k], at[k], acc);
            out[((size_t)b * CD + c) * ND + n0 + n] = acc;
        }
    }
}

extern "C" void kernel_launch(void* const* d_in, const int* in_sizes, int n_in,
                              void* d_out, int out_size, void* d_ws, size_t ws_size,
                              hipStream_t stream)
{
    const float* x    = (const float*)d_in[0];   // (8, 64, 4096, 20) f32
    const float* W    = (const float*)d_in[1];   // (1, 128) f32
    const float* bias = (const float*)d_in[2];   // (1,) f32
    float*       out  = (float*)d_out;           // (8, 64, 4096, 1) f32

    const int blocks = BD * (ND / NB);           // 2048
    graph_attn_wmma_kernel<<<blocks, THREADS, 0, stream>>>(x, W, bias, out);
}